// GraphSAGEEncoder_66099546686043
// MI455X (gfx1250) — compile-verified
//
#include <hip/hip_runtime.h>
#include <hip/hip_bf16.h>

// GraphSAGE 2-layer encoder for MI455X (gfx1250).
// Scatter phase is L2-atomic bound (~1 GB of edge traffic, whole working set
// fits in 192MB L2); dense linear layers use V_WMMA_F32_16X16X4_F32 so the
// GEMM path runs on the matrix pipe at full fp32 precision.

#define NUM_NODES 50000
#define NUM_EDGES 640000
#define FEAT_DIM  64
#define HIDDEN    128
#define OUT_DIM   128

typedef __attribute__((ext_vector_type(2))) float v2f;
typedef __attribute__((ext_vector_type(8))) float v8f;

// ---------------------------------------------------------------------------
// Degree count: one thread per edge.
// ---------------------------------------------------------------------------
__global__ void deg_kernel(const int* __restrict__ dst, float* __restrict__ deg,
                           int E) {
    int e = blockIdx.x * blockDim.x + threadIdx.x;
    if (e < E) atomicAdd(&deg[dst[e]], 1.0f);
}

// ---------------------------------------------------------------------------
// Scatter-add of source features into agg[dst]. One thread per (edge, feat):
// for a fixed edge, D consecutive threads cover the feature dim, so both the
// gather load x[s*D+f] and the atomic target agg[d*D+f] are coalesced.
// ---------------------------------------------------------------------------
__global__ void scatter_add_kernel(const int* __restrict__ src,
                                   const int* __restrict__ dst,
                                   const float* __restrict__ x,
                                   float* __restrict__ agg,
                                   int E, int D) {
    int idx = blockIdx.x * blockDim.x + threadIdx.x;
    int total = E * D;                 // <= 81.9M, fits int32
    if (idx >= total) return;
    int e = idx / D;
    int f = idx - e * D;
    int s = src[e];
    int d = dst[e];
    atomicAdd(&agg[d * D + f], x[s * D + f]);
}

// ---------------------------------------------------------------------------
// Fused SAGE linear: out = act( (agg/deg) @ Wl.T + bl + x @ Wr.T )
// One wave produces one 16x16 output tile via V_WMMA_F32_16X16X4_F32,
// accumulating both GEMMs into the same 8-VGPR accumulator.
//
// A 16x4 f32 layout: lanes 0-15 hold K={k0,k0+1}, lanes 16-31 hold K={k0+2,k0+3}.
// B 4x16 f32 layout mirrors A with N striped across lanes (B = W.T).
// C/D 16x16 f32: VGPR r, lanes<16 -> (M=r, N=lane); lanes>=16 -> (M=r+8, N=lane-16).
// ---------------------------------------------------------------------------
template <int D>
__global__ void sage_linear_wmma_kernel(const float* __restrict__ agg,
                                        const float* __restrict__ deg,
                                        const float* __restrict__ x,
                                        const float* __restrict__ Wl,  // [OUT, D]
                                        const float* __restrict__ bl,  // [OUT]
                                        const float* __restrict__ Wr,  // [OUT, D]
                                        float* __restrict__ out,       // [N, OUT]
                                        int N, int OUT, int relu) {
    const int lane   = threadIdx.x & 31;
    const int waveId = (blockIdx.x * blockDim.x + threadIdx.x) >> 5;

    const int tilesN = OUT >> 4;                 // 16-wide column tiles
    const int tileM  = (waveId / tilesN) << 4;   // node-row tile
    const int tileN  = (waveId % tilesN) << 4;   // out-channel tile
    if (tileM >= N) return;                      // wave-uniform; EXEC stays full

    const int laneLo = lane & 15;
    const int laneHi = lane >> 4;                // selects K pair {0,1} vs {2,3}
    const int rowA   = tileM + laneLo;           // A-matrix row for this lane
    const int colB   = tileN + laneLo;           // B-matrix col (output channel)

    const float dv     = deg[rowA];
    const float invdeg = 1.0f / fmaxf(dv, 1.0f);

    const float* __restrict__ aRow = agg + rowA * D;
    const float* __restrict__ xRow = x   + rowA * D;
    const float* __restrict__ wlRow = Wl + colB * D;
    const float* __restrict__ wrRow = Wr + colB * D;

    v8f acc = {};

    // Pass 1: mean @ Wl.T
#pragma unroll 4
    for (int k0 = 0; k0 < D; k0 += 4) {
        const int ka = k0 + (laneHi << 1);
        v2f a, b;
        a.x = aRow[ka]     * invdeg;
        a.y = aRow[ka + 1] * invdeg;
        b.x = wlRow[ka];
        b.y = wlRow[ka + 1];
        acc = __builtin_amdgcn_wmma_f32_16x16x4_f32(
            /*neg_a=*/false, a, /*neg_b=*/false, b,
            /*c_mod=*/(short)0, acc, /*reuse_a=*/false, /*reuse_b=*/false);
    }
    // Pass 2: x @ Wr.T
#pragma unroll 4
    for (int k0 = 0; k0 < D; k0 += 4) {
        const int ka = k0 + (laneHi << 1);
        v2f a, b;
        a.x = xRow[ka];
        a.y = xRow[ka + 1];
        b.x = wrRow[ka];
        b.y = wrRow[ka + 1];
        acc = __builtin_amdgcn_wmma_f32_16x16x4_f32(
            false, a, false, b, (short)0, acc, false, false);
    }

    const float bias = bl[colB];
#pragma unroll
    for (int r = 0; r < 8; ++r) {
        const int row = tileM + r + (laneHi << 3);
        float v = acc[r] + bias;
        if (relu) v = fmaxf(v, 0.0f);
        out[row * OUT + (tileN + laneLo)] = v;
    }
}

// ---------------------------------------------------------------------------
// Launch: memset scratch -> deg -> scatter1 -> linear1(relu) -> scatter2 ->
// linear2. All on `stream`, graph-capture safe (hipMemsetAsync only).
// ---------------------------------------------------------------------------
extern "C" void kernel_launch(void* const* d_in, const int* in_sizes, int n_in,
                              void* d_out, int out_size, void* d_ws, size_t ws_size,
                              hipStream_t stream) {
    const int   N = NUM_NODES;
    const int   E = NUM_EDGES;

    const int*   edge = (const int*)d_in[0];
    const int*   src  = edge;
    const int*   dst  = edge + E;
    const float* emb  = (const float*)d_in[1];
    const float* W1l  = (const float*)d_in[2];
    const float* b1l  = (const float*)d_in[3];
    const float* W1r  = (const float*)d_in[4];
    const float* W2l  = (const float*)d_in[5];
    const float* b2l  = (const float*)d_in[6];
    const float* W2r  = (const float*)d_in[7];
    float*       out  = (float*)d_out;

    // Workspace layout (floats): agg1[N*64] | deg[N] | agg2[N*128] | h[N*128]
    float* ws   = (float*)d_ws;
    float* agg1 = ws;
    float* deg  = ws + (size_t)N * FEAT_DIM;
    float* agg2 = ws + (size_t)N * (FEAT_DIM + 1);
    float* h    = ws + (size_t)N * (FEAT_DIM + 1 + HIDDEN);

    // Zero agg1 + deg + agg2 in one contiguous memset (h is fully overwritten).
    hipMemsetAsync(d_ws, 0,
                   (size_t)N * (FEAT_DIM + 1 + HIDDEN) * sizeof(float), stream);

    // Degree
    deg_kernel<<<(E + 255) / 256, 256, 0, stream>>>(dst, deg, E);

    // Layer 1 scatter: E*64 threads
    {
        int total = E * FEAT_DIM;
        scatter_add_kernel<<<(total + 255) / 256, 256, 0, stream>>>(
            src, dst, emb, agg1, E, FEAT_DIM);
    }

    // Layer 1 linear (WMMA) + ReLU -> h
    {
        int waves  = (N / 16) * (HIDDEN / 16);   // 3125 * 8 = 25000
        int blocks = waves / 4;                  // 128 threads = 4 waves/block
        sage_linear_wmma_kernel<FEAT_DIM><<<blocks, 128, 0, stream>>>(
            agg1, deg, emb, W1l, b1l, W1r, h, N, HIDDEN, /*relu=*/1);
    }

    // Layer 2 scatter: E*128 threads
    {
        int total = E * HIDDEN;
        scatter_add_kernel<<<(total + 255) / 256, 256, 0, stream>>>(
            src, dst, h, agg2, E, HIDDEN);
    }

    // Layer 2 linear (WMMA), no activation -> d_out
    {
        int waves  = (N / 16) * (OUT_DIM / 16);
        int blocks = waves / 4;
        sage_linear_wmma_kernel<HIDDEN><<<blocks, 128, 0, stream>>>(
            agg2, deg, h, W2l, b2l, W2r, out, N, OUT_DIM, /*relu=*/0);
    }
}